// SelfAttention_7730941133365
// MI455X (gfx1250) — compile-verified
//
#include <hip/hip_runtime.h>

// ---------------------------------------------------------------------------
// Banded self-attention forward for MI455X (gfx1250), fp32 via
// V_WMMA_F32_16X16X4_F32 + async global->LDS (ASYNCcnt) double-buffered GEMM.
// Outputs: out [S,E] then masked scores [H,S,S].
// ---------------------------------------------------------------------------

typedef float v2f __attribute__((ext_vector_type(2)));
typedef float v8f __attribute__((ext_vector_type(8)));

#define S_LEN 2048
#define E_DIM 1024
#define H_NUM 16
#define DH_DIM 64
#define KNB 64
#define MASKVAL (-1e10f)

static __device__ __forceinline__ v8f wmma_f32(v2f a, v2f b, v8f c) {
    return __builtin_amdgcn_wmma_f32_16x16x4_f32(
        /*neg_a=*/false, a, /*neg_b=*/false, b,
        /*c_mod=*/(short)0, c, /*reuse_a=*/false, /*reuse_b=*/false);
}

// Async 16-byte global -> LDS copy (CDNA5, tracked by ASYNCcnt).
// vdst VGPR carries the wave-relative LDS byte address (low 32 bits of the
// flat shared pointer), addr is the 64-bit global address.
static __device__ __forceinline__ void async_copy16(float* lds, const float* g) {
    unsigned off = (unsigned)(unsigned long long)lds;
    asm volatile("global_load_async_to_lds_b128 %0, %1, off"
                 :: "v"(off), "v"(g)
                 : "memory");
}
static __device__ __forceinline__ void wait_asynccnt_le3() {
    asm volatile("s_wait_asynccnt 0x3" ::: "memory");
}
static __device__ __forceinline__ void wait_asynccnt_0() {
    asm volatile("s_wait_asynccnt 0x0" ::: "memory");
}

// ---------------------------------------------------------------------------
// Kernel 1: fill the full scores tensor with MASKVAL (bandwidth-bound).
// ---------------------------------------------------------------------------
__global__ void fill_scores_kernel(float4* __restrict__ p, long long n4) {
    long long i = (long long)blockIdx.x * blockDim.x + threadIdx.x;
    long long stride = (long long)gridDim.x * blockDim.x;
    float4 v = make_float4(MASKVAL, MASKVAL, MASKVAL, MASKVAL);
    for (long long k = i; k < n4; k += stride) p[k] = v;
}

// ---------------------------------------------------------------------------
// Kernel 2/4: C[M,N] = A[M,K] @ B[N,K]^T + bias[N], fp32 WMMA.
// Block: 256 threads = 8 waves (4 in M x 2 in N). Block tile 128x64,
// wave tile 32x32 (2x2 accumulators of 16x16). K slab = 16.
// Double-buffered LDS filled by async B128 copies (3 per thread per slab).
// Requires M%128==0, N%64==0, K%16==0 (true here).
// ---------------------------------------------------------------------------
#define GM_BM 128
#define GM_BN 64
#define GM_KC 16
#define GM_LDSP 20   // row stride (floats): 80B -> every row 16B aligned

__global__ __launch_bounds__(256) void gemm_nt_bias_kernel(
    const float* __restrict__ A, const float* __restrict__ B,
    const float* __restrict__ bias, float* __restrict__ C,
    int M, int N, int K) {
    __shared__ float As[2][GM_BM * GM_LDSP];
    __shared__ float Bs[2][GM_BN * GM_LDSP];

    const int t = threadIdx.x;
    const int lane = t & 31;
    const int wid = t >> 5;          // 0..7
    const int wm = wid & 3;          // wave row (M)
    const int wn = wid >> 2;         // wave col (N)
    const int l16 = lane & 15;
    const int lh = lane >> 4;        // lane half (0/1)

    const int Mblk = blockIdx.y * GM_BM;
    const int Nblk = blockIdx.x * GM_BN;

    // Per-thread async copy coordinates (fixed across slabs).
    const int ar0 = t >> 2;                 // A chunk 0 row (0..63)
    const int ar1 = (t + 256) >> 2;         // A chunk 1 row (64..127)
    const int ac4 = (t & 3) << 2;           // k-subcolumn (0,4,8,12)
    const int br  = t >> 2;                 // B row (0..63)

    v8f acc[2][2] = {};

    const int NK = K / GM_KC;

    // ---- issue async loads for a K-slab into LDS buffer `buf` ----
    auto issue_slab = [&](int buf, int k0) {
        async_copy16(&As[buf][ar0 * GM_LDSP + ac4],
                     A + (size_t)(Mblk + ar0) * K + k0 + ac4);
        async_copy16(&As[buf][ar1 * GM_LDSP + ac4],
                     A + (size_t)(Mblk + ar1) * K + k0 + ac4);
        async_copy16(&Bs[buf][br * GM_LDSP + ac4],
                     B + (size_t)(Nblk + br) * K + k0 + ac4);
    };

    issue_slab(0, 0);

    for (int ko = 0; ko < NK; ++ko) {
        const int cur = ko & 1;
        const bool more = (ko + 1 < NK);
        if (more) issue_slab(cur ^ 1, (ko + 1) * GM_KC);

        // Wait for this slab's 3 copies (leave next slab's 3 in flight).
        if (more) wait_asynccnt_le3(); else wait_asynccnt_0();
        __syncthreads();

        // --- 4 WMMA K-steps of 4 over the 16-wide slab ---
        #pragma unroll
        for (int kk = 0; kk < 4; ++kk) {
            const int koff = 4 * kk + lh * 2;
            v2f aF[2], bF[2];
            #pragma unroll
            for (int mt = 0; mt < 2; ++mt) {
                int row = wm * 32 + mt * 16 + l16;
                aF[mt] = *(const v2f*)&As[cur][row * GM_LDSP + koff];
            }
            #pragma unroll
            for (int nt = 0; nt < 2; ++nt) {
                int row = wn * 32 + nt * 16 + l16;
                bF[nt] = *(const v2f*)&Bs[cur][row * GM_LDSP + koff];
            }
            #pragma unroll
            for (int mt = 0; mt < 2; ++mt)
                #pragma unroll
                for (int nt = 0; nt < 2; ++nt)
                    acc[mt][nt] = wmma_f32(aF[mt], bF[nt], acc[mt][nt]);
        }
        __syncthreads();   // all waves done reading buf `cur`
    }

    // --- epilogue: bias + store ---
    #pragma unroll
    for (int nt = 0; nt < 2; ++nt) {
        int col = Nblk + wn * 32 + nt * 16 + l16;
        float bv = bias[col];
        #pragma unroll
        for (int mt = 0; mt < 2; ++mt) {
            int rowbase = Mblk + wm * 32 + mt * 16 + lh * 8;
            #pragma unroll
            for (int r = 0; r < 8; ++r)
                C[(size_t)(rowbase + r) * N + col] = acc[mt][nt][r] + bv;
        }
    }
}

// ---------------------------------------------------------------------------
// Kernel 3: banded attention. One wave (32 threads) per (head, 16-query tile).
//   scores tile = (q @ k^T)*scale with |i-j|<=64 mask -> stored to scores[]
//   online softmax across band column tiles, flash-style rescale of P@V accs.
// qkv layout: [S, 3E]; q at col h*64+d, k at E + h*64+d, v at 2E + h*64+d.
// ---------------------------------------------------------------------------
#define PSTR 18   // P-tile row stride (even -> 8B-aligned b64 fragment loads)

__global__ __launch_bounds__(32) void banded_attn_kernel(
    const float* __restrict__ qkv, float* __restrict__ scores,
    float* __restrict__ ctx) {
    __shared__ float P[16 * PSTR];   // probability tile staging (C -> A layout)

    const int lane = threadIdx.x;
    const int l16 = lane & 15;
    const int lh = lane >> 4;

    const int h = blockIdx.x >> 7;        // / (S/16)
    const int qt = blockIdx.x & 127;
    const int i0 = qt << 4;
    const float scale = 0.125f;           // DH^-0.5

    const size_t R3E = 3 * E_DIM;
    const float* qsrc = qkv + (size_t)(i0 + l16) * R3E + h * DH_DIM;
    const float* ksrc = qkv + E_DIM + h * DH_DIM;
    const float* vsrc = qkv + 2 * E_DIM + h * DH_DIM;
    float* srow = scores + (size_t)h * S_LEN * S_LEN;

    // Preload the 16 q A-fragments (K = DH = 64, 4 per step).
    v2f qf[16];
    #pragma unroll
    for (int kk = 0; kk < 16; ++kk)
        qf[kk] = *(const v2f*)(qsrc + 4 * kk + lh * 2);

    v8f oacc[4] = {};
    float mrun[8], lrun[8];
    #pragma unroll
    for (int r = 0; r < 8; ++r) { mrun[r] = -1e30f; lrun[r] = 0.0f; }

    const int jt_lo = (qt >= 4) ? (qt - 4) : 0;
    const int jt_hi = (qt <= 123) ? (qt + 4) : 127;

    for (int jt = jt_lo; jt <= jt_hi; ++jt) {
        const int j0 = jt << 4;

        // ---- scores tile: 16 WMMA steps over DH=64 ----
        v8f sacc = {};
        const float* kbase = ksrc + (size_t)(j0 + l16) * R3E;
        #pragma unroll
        for (int kk = 0; kk < 16; ++kk) {
            v2f bf = *(const v2f*)(kbase + 4 * kk + lh * 2);
            sacc = wmma_f32(qf[kk], bf, sacc);
        }

        // ---- mask + store + online softmax ----
        const int rowbase = i0 + lh * 8;
        const int col = j0 + l16;
        #pragma unroll
        for (int r = 0; r < 8; ++r) {
            int row = rowbase + r;
            float sv = sacc[r] * scale;
            int d = row - col; d = (d < 0) ? -d : d;
            sv = (d <= KNB) ? sv : MASKVAL;
            srow[(size_t)row * S_LEN + col] = sv;

            // row max over the 16-lane group
            float v = sv;
            #pragma unroll
            for (int msk = 1; msk < 16; msk <<= 1)
                v = fmaxf(v, __shfl_xor(v, msk, 32));
            float mnew = fmaxf(mrun[r], v);
            float alpha = __expf(mrun[r] - mnew);
            mrun[r] = mnew;

            float p = __expf(sv - mnew);
            float ps = p;
            #pragma unroll
            for (int msk = 1; msk < 16; msk <<= 1)
                ps += __shfl_xor(ps, msk, 32);
            lrun[r] = lrun[r] * alpha + ps;

            #pragma unroll
            for (int nt = 0; nt < 4; ++nt)
                oacc[nt][r] *= alpha;

            P[(lh * 8 + r) * PSTR + l16] = p;   // C-layout -> LDS
        }
        __syncthreads();

        // ---- P (16x16) @ V (16x64): 4 K-steps x 4 d-tiles ----
        #pragma unroll
        for (int kk2 = 0; kk2 < 4; ++kk2) {
            v2f af = *(const v2f*)&P[l16 * PSTR + 4 * kk2 + lh * 2];
            const int jrow = j0 + 4 * kk2 + lh * 2;
            const float* v0 = vsrc + (size_t)jrow * R3E;
            const float* v1 = v0 + R3E;
            #pragma unroll
            for (int nt = 0; nt < 4; ++nt) {
                int dc = nt * 16 + l16;
                v2f bf;
                bf.x = v0[dc];
                bf.y = v1[dc];
                oacc[nt] = wmma_f32(af, bf, oacc[nt]);
            }
        }
        __syncthreads();
    }

    // ---- normalize and store context ----
    #pragma unroll
    for (int nt = 0; nt < 4; ++nt) {
        int dc = h * DH_DIM + nt * 16 + l16;
        #pragma unroll
        for (int r = 0; r < 8; ++r) {
            int row = i0 + lh * 8 + r;
            ctx[(size_t)row * E_DIM + dc] = oacc[nt][r] / lrun[r];
        }
    }
}

// ---------------------------------------------------------------------------
extern "C" void kernel_launch(void* const* d_in, const int* in_sizes, int n_in,
                              void* d_out, int out_size, void* d_ws, size_t ws_size,
                              hipStream_t stream) {
    (void)in_sizes; (void)n_in; (void)out_size; (void)ws_size;
    const float* x    = (const float*)d_in[0];
    const float* Wqkv = (const float*)d_in[1];
    const float* bqkv = (const float*)d_in[2];
    const float* Wout = (const float*)d_in[3];
    const float* bout = (const float*)d_in[4];

    float* out    = (float*)d_out;
    float* scores = out + (size_t)S_LEN * E_DIM;

    float* qkv = (float*)d_ws;                       // [S, 3E]
    float* ctx = qkv + (size_t)S_LEN * 3 * E_DIM;    // [S, E]

    // 1) fill scores with MASKVAL
    long long n4 = (long long)H_NUM * S_LEN * S_LEN / 4;
    fill_scores_kernel<<<4096, 256, 0, stream>>>((float4*)scores, n4);

    // 2) qkv = x @ W_qkv^T + b_qkv   [2048, 3072]
    gemm_nt_bias_kernel<<<dim3(3 * E_DIM / GM_BN, S_LEN / GM_BM), 256, 0, stream>>>(
        x, Wqkv, bqkv, qkv, S_LEN, 3 * E_DIM, E_DIM);

    // 3) banded attention: scores (masked) + ctx
    banded_attn_kernel<<<H_NUM * (S_LEN / 16), 32, 0, stream>>>(qkv, scores, ctx);

    // 4) out = ctx @ W_out^T + b_out  [2048, 1024]
    gemm_nt_bias_kernel<<<dim3(E_DIM / GM_BN, S_LEN / GM_BM), 256, 0, stream>>>(
        ctx, Wout, bout, out, S_LEN, E_DIM, E_DIM);
}